// VN_Ori_Globa_22067541967081
// MI455X (gfx1250) — compile-verified
//
#include <hip/hip_runtime.h>
#include <hip/hip_bf16.h>

// ---------------------------------------------------------------------------
// VN-DGCNN forward for MI455X (gfx1250, wave32).
//   - kNN Gram matrices + conv6 GEMM use V_WMMA_F32_16X16X4_F32 (fp32 WMMA).
//   - All K dims are == 3 (mod 4): fragment .x loads are always in-bounds and
//     only the tail .y needs a value-select -> branch-free WMMA inner loops.
//   - Top-20 selection is register-resident with a branch-free unrolled
//     sorted-insert (v_cmp/v_cndmask network, no divergent while loops).
// ---------------------------------------------------------------------------

#define EPSV   1e-6f
#define BN_EPS 1e-5f
#define SLOPE  0.2f
#define KNN    20
#define NPTS   2048
#define NB     8
#define FDIM   341

typedef __attribute__((ext_vector_type(2))) float v2f;
typedef __attribute__((ext_vector_type(8))) float v8f;

#if defined(__gfx1250__) && __has_builtin(__builtin_amdgcn_wmma_f32_16x16x4_f32)
#define WMMA4(acc, a, b)                                                        \
  acc = __builtin_amdgcn_wmma_f32_16x16x4_f32(false, (a), false, (b),           \
                                              (short)0, (acc), false, false)
#else
// Host-pass / fallback placeholder (keeps sema happy; histogram shows wmma=0).
#define WMMA4(acc, a, b) do { acc[0] += (a).x * (b).x; } while (0)
#endif

// ---------------------------------------------------------------------------
// VN BatchNorm (eval) + VN LeakyReLU on a 3-vector, direction q.
// ---------------------------------------------------------------------------
__device__ inline void vn_bn_leaky(float& p0, float& p1, float& p2,
                                   float q0, float q1, float q2,
                                   float g, float be, float mu, float va) {
  float nrm = sqrtf(p0*p0 + p1*p1 + p2*p2) + EPSV;
  float nb  = g * (nrm - mu) * rsqrtf(va + BN_EPS) + be;
  float s   = nb / nrm;
  p0 *= s; p1 *= s; p2 *= s;
  float dot = p0*q0 + p1*q1 + p2*q2;
  if (dot < 0.f) {
    float t = dot / (q0*q0 + q1*q1 + q2*q2 + EPSV);
    p0 = SLOPE * p0 + (1.f - SLOPE) * (p0 - t * q0);
    p1 = SLOPE * p1 + (1.f - SLOPE) * (p1 - t * q1);
    p2 = SLOPE * p2 + (1.f - SLOPE) * (p2 - t * q2);
  }
}

__device__ inline void vn_leaky_only(float& p0, float& p1, float& p2,
                                     float q0, float q1, float q2) {
  float dot = p0*q0 + p1*q1 + p2*q2;
  if (dot < 0.f) {
    float t = dot / (q0*q0 + q1*q1 + q2*q2 + EPSV);
    p0 = SLOPE * p0 + (1.f - SLOPE) * (p0 - t * q0);
    p1 = SLOPE * p1 + (1.f - SLOPE) * (p1 - t * q1);
    p2 = SLOPE * p2 + (1.f - SLOPE) * (p2 - t * q2);
  }
}

// ---------------------------------------------------------------------------
// Fused kNN: WMMA f32 Gram tiles + incremental per-row top-20.
// One wave (32 lanes) per (batch, 16-row tile). X is [D rows][NPTS cols].
// Lanes 0..15 each own one row; their top-20 lists live in registers and are
// updated with a fully unrolled, branch-free sorted insert.
// ---------------------------------------------------------------------------
template <int D>
__global__ __launch_bounds__(32)
void knn_wmma_kernel(const float* __restrict__ X, int batchStride,
                     int* __restrict__ idx_out) {
  static_assert(D % 4 == 3, "tail handling assumes D % 4 == 3");
  constexpr int KS = (D + 3) / 4;
  const int N      = NPTS;
  const int b      = blockIdx.y;
  const int mbase  = blockIdx.x * 16;
  const int lane   = threadIdx.x;
  const int lane15 = lane & 15;
  const int khalf  = (lane >= 16) ? 2 : 0;
  const float* Xb  = X + (size_t)b * batchStride;

  __shared__ float xxs[NPTS];
  __shared__ float dist[16][17];

  // Column squared norms for this batch, staged once in LDS.
  for (int n = lane; n < N; n += 32) {
    float s = 0.f;
#pragma unroll
    for (int d = 0; d < D; ++d) { float v = Xb[(size_t)d * N + n]; s += v * v; }
    xxs[n] = s;
  }
  __syncthreads();

  // Register-resident top-20 (valid in lanes 0..15; statically indexed only).
  float tv[KNN];
  int   ti[KNN];
#pragma unroll
  for (int t = 0; t < KNN; ++t) { tv[t] = -3.0e38f; ti[t] = 0; }

  // A fragments (16x4 f32 per k-step). .x is always in-bounds (k0 <= D-1);
  // .y uses a clamped load + value select (no exec-mask predication).
  v2f afrag[KS];
  const int row = mbase + lane15;
#pragma unroll
  for (int kk = 0; kk < KS; ++kk) {
    const int  k0 = kk * 4 + khalf;
    const int  k1 = k0 + 1;
    const bool ok = (k1 < D);
    v2f a;
    a.x = Xb[(size_t)k0 * N + row];
    float ay = Xb[(size_t)(ok ? k1 : (D - 1)) * N + row];
    a.y = ok ? ay : 0.f;
    afrag[kk] = a;
  }

  const int   rofs = (lane >= 16) ? 8 : 0;
  const float rx   = xxs[mbase + lane15];     // row norm (used by lanes 0..15)
  for (int jt = 0; jt < N / 16; ++jt) {
    const int col = jt * 16 + lane15;
    if (jt + 1 < N / 16)
      __builtin_prefetch(Xb + (size_t)khalf * N + col + 16);   // global_prefetch_b8
    v8f acc = {0.f, 0.f, 0.f, 0.f, 0.f, 0.f, 0.f, 0.f};
#pragma unroll
    for (int kk = 0; kk < KS; ++kk) {
      const int  k0 = kk * 4 + khalf;
      const int  k1 = k0 + 1;
      const bool ok = (k1 < D);
      v2f bf;
      bf.x = Xb[(size_t)k0 * N + col];
      float by = Xb[(size_t)(ok ? k1 : (D - 1)) * N + col];
      bf.y = ok ? by : 0.f;
      WMMA4(acc, afrag[kk], bf);
    }
    float xxc = xxs[col];
#pragma unroll
    for (int v = 0; v < 8; ++v)
      dist[v + rofs][lane15] = 2.f * acc[v] - xxc;   // 2G - xx_col
    __syncthreads();
    if (lane < 16) {
#pragma unroll 1
      for (int cc = 0; cc < 16; ++cc) {
        float val = dist[lane][cc] - rx;             // neg squared distance
        if (val > tv[KNN - 1]) {                     // rare (~k*ln(N)/N)
          const int vi = jt * 16 + cc;
          // Branch-free sorted insert (descending), fully unrolled:
          //   new[p] = old[p]              if old[p]   >= val
          //          = old[p-1]            if old[p-1] <  val   (shift down)
          //          = val                 otherwise            (insert here)
#pragma unroll
          for (int p = KNN - 1; p >= 1; --p) {
            bool keep  = (tv[p] >= val);
            bool shift = (tv[p - 1] < val);
            float nv = keep ? tv[p] : (shift ? tv[p - 1] : val);
            int   ni = keep ? ti[p] : (shift ? ti[p - 1] : vi);
            tv[p] = nv; ti[p] = ni;
          }
          bool keep0 = (tv[0] >= val);
          tv[0] = keep0 ? tv[0] : val;
          ti[0] = keep0 ? ti[0] : vi;
        }
      }
    }
    __syncthreads();
  }
  if (lane < 16) {
    int* op = idx_out + ((size_t)b * NPTS + (mbase + lane)) * KNN;
#pragma unroll
    for (int t = 0; t < KNN; ++t) op[t] = ti[t];
  }
}

// ---------------------------------------------------------------------------
// Batched GEMM with WMMA f32: Out[b][M][Nc] = A[M][KD] * Xin[b][KD][Nc].
// One wave computes NT adjacent 16x16 column tiles, reusing the A fragments
// held in VGPRs (conv6: M=341, KD=63, Nc=3*2048, NT=4 -> 64 WMMAs/wave).
// ---------------------------------------------------------------------------
template <int KD, int NT>
__global__ __launch_bounds__(32)
void gemm_wmma_kernel(const float* __restrict__ A, const float* __restrict__ Xin,
                      float* __restrict__ Out, int M, int Nc) {
  static_assert(KD % 4 == 3, "tail handling assumes KD % 4 == 3");
  constexpr int KS = (KD + 3) / 4;
  const int b      = blockIdx.z;
  const int mbase  = blockIdx.x * 16;
  const int nbase0 = blockIdx.y * (16 * NT);
  const int lane   = threadIdx.x;
  const int lane15 = lane & 15;
  const int khalf  = (lane >= 16) ? 2 : 0;
  const float* Xb  = Xin + (size_t)b * KD * Nc;

  const int row  = mbase + lane15;
  const int rowc = (row < M) ? row : (M - 1);    // clamp for loads; store guards
  v2f af[KS];
#pragma unroll
  for (int kk = 0; kk < KS; ++kk) {
    const int  k0 = kk * 4 + khalf;
    const int  k1 = k0 + 1;
    const bool ok = (k1 < KD);
    v2f a;
    a.x = A[(size_t)rowc * KD + k0];
    float ay = A[(size_t)rowc * KD + (ok ? k1 : (KD - 1))];
    a.y = ok ? ay : 0.f;
    af[kk] = a;
  }

  const int rofs = (lane >= 16) ? 8 : 0;
  float* Ob = Out + (size_t)b * M * Nc;
#pragma unroll
  for (int t = 0; t < NT; ++t) {
    const int col = nbase0 + t * 16 + lane15;
    v8f acc = {0.f, 0.f, 0.f, 0.f, 0.f, 0.f, 0.f, 0.f};
#pragma unroll
    for (int kk = 0; kk < KS; ++kk) {
      const int  k0 = kk * 4 + khalf;
      const int  k1 = k0 + 1;
      const bool ok = (k1 < KD);
      v2f bf;
      bf.x = Xb[(size_t)k0 * Nc + col];
      float by = Xb[(size_t)(ok ? k1 : (KD - 1)) * Nc + col];
      bf.y = ok ? by : 0.f;
      WMMA4(acc, af[kk], bf);
    }
#pragma unroll
    for (int v = 0; v < 8; ++v) {
      int r = mbase + v + rofs;
      if (r < M) Ob[(size_t)r * Nc + col] = acc[v];
    }
  }
}

// ---------------------------------------------------------------------------
// Gather edge features (nbr-ctr | ctr) and apply a VN linear+BN+leaky block.
// src: [B][C][3][NPTS] region (batch stride given); out: [B][OC][3][NPTS][KNN].
// ---------------------------------------------------------------------------
__global__ void vn_edgeconv_kernel(const float* __restrict__ src, int srcBatchStride, int C,
                                   const int* __restrict__ idx,
                                   const float* __restrict__ Wf, const float* __restrict__ Wd,
                                   const float* __restrict__ g, const float* __restrict__ be,
                                   const float* __restrict__ mu, const float* __restrict__ va,
                                   int OC, float* __restrict__ out, int total) {
  int tid = blockIdx.x * blockDim.x + threadIdx.x;
  if (tid >= total) return;
  const int N = NPTS;
  int oc = tid % OC;
  int kk = (tid / OC) % KNN;
  int n  = (tid / (OC * KNN)) % N;
  int b  = tid / (OC * KNN * N);
  const float* sb = src + (size_t)b * srcBatchStride;
  int j = idx[((size_t)b * N + n) * KNN + kk];
  float p0 = 0, p1 = 0, p2 = 0, q0 = 0, q1 = 0, q2 = 0;
  const int C2 = 2 * C;
  for (int c = 0; c < C2; ++c) {
    float wf = Wf[oc * C2 + c], wd = Wd[oc * C2 + c];
    float e0, e1, e2;
    if (c < C) {
      const float* sc = sb + (size_t)c * 3 * N;
      e0 = sc[j] - sc[n];
      e1 = sc[N + j] - sc[N + n];
      e2 = sc[2 * N + j] - sc[2 * N + n];
    } else {
      const float* sc = sb + (size_t)(c - C) * 3 * N;
      e0 = sc[n]; e1 = sc[N + n]; e2 = sc[2 * N + n];
    }
    p0 += wf * e0; p1 += wf * e1; p2 += wf * e2;
    q0 += wd * e0; q1 += wd * e1; q2 += wd * e2;
  }
  vn_bn_leaky(p0, p1, p2, q0, q1, q2, g[oc], be[oc], mu[oc], va[oc]);
  size_t ob = ((size_t)b * OC + oc) * 3 * N * KNN + (size_t)n * KNN + kk;
  out[ob]                        = p0;
  out[ob + (size_t)N * KNN]      = p1;
  out[ob + 2 * (size_t)N * KNN]  = p2;
}

// ---------------------------------------------------------------------------
// Plain VN conv on [B][C][3][N][K] fused with mean over K -> writes into a
// channel slice of x123 ([B][outChanTot][3][N]).
// ---------------------------------------------------------------------------
__global__ void vn_conv_mean_kernel(const float* __restrict__ src, int C,
                                    const float* __restrict__ Wf, const float* __restrict__ Wd,
                                    const float* __restrict__ g, const float* __restrict__ be,
                                    const float* __restrict__ mu, const float* __restrict__ va,
                                    int OC, float* __restrict__ out, int outChanOfs,
                                    int outChanTot, int total) {
  int tid = blockIdx.x * blockDim.x + threadIdx.x;
  if (tid >= total) return;
  const int N = NPTS;
  int oc = tid % OC;
  int n  = (tid / OC) % N;
  int b  = tid / (OC * N);
  const float* sb = src + (size_t)b * C * 3 * N * KNN;
  float m0 = 0.f, m1 = 0.f, m2 = 0.f;
  for (int kk = 0; kk < KNN; ++kk) {
    float p0 = 0, p1 = 0, p2 = 0, q0 = 0, q1 = 0, q2 = 0;
    for (int c = 0; c < C; ++c) {
      float wf = Wf[oc * C + c], wd = Wd[oc * C + c];
      size_t e = ((size_t)c * 3 * N + n) * KNN + kk;
      float e0 = sb[e];
      float e1 = sb[e + (size_t)N * KNN];
      float e2 = sb[e + 2 * (size_t)N * KNN];
      p0 += wf * e0; p1 += wf * e1; p2 += wf * e2;
      q0 += wd * e0; q1 += wd * e1; q2 += wd * e2;
    }
    vn_bn_leaky(p0, p1, p2, q0, q1, q2, g[oc], be[oc], mu[oc], va[oc]);
    m0 += p0; m1 += p1; m2 += p2;
  }
  const float inv = 1.f / (float)KNN;
  m0 *= inv; m1 *= inv; m2 *= inv;
  size_t ob = ((size_t)b * outChanTot + outChanOfs + oc) * 3 * N + n;
  out[ob] = m0; out[ob + N] = m1; out[ob + 2 * N] = m2;
}

// ---------------------------------------------------------------------------
// Gather edge conv fused with mean over K (conv5 path).
// ---------------------------------------------------------------------------
__global__ void vn_edgeconv_mean_kernel(const float* __restrict__ src, int srcBatchStride, int C,
                                        const int* __restrict__ idx,
                                        const float* __restrict__ Wf, const float* __restrict__ Wd,
                                        const float* __restrict__ g, const float* __restrict__ be,
                                        const float* __restrict__ mu, const float* __restrict__ va,
                                        int OC, float* __restrict__ out, int outChanOfs,
                                        int outChanTot, int total) {
  int tid = blockIdx.x * blockDim.x + threadIdx.x;
  if (tid >= total) return;
  const int N = NPTS;
  int oc = tid % OC;
  int n  = (tid / OC) % N;
  int b  = tid / (OC * N);
  const float* sb = src + (size_t)b * srcBatchStride;
  const int C2 = 2 * C;
  float m0 = 0.f, m1 = 0.f, m2 = 0.f;
  for (int kk = 0; kk < KNN; ++kk) {
    int j = idx[((size_t)b * N + n) * KNN + kk];
    float p0 = 0, p1 = 0, p2 = 0, q0 = 0, q1 = 0, q2 = 0;
    for (int c = 0; c < C2; ++c) {
      float wf = Wf[oc * C2 + c], wd = Wd[oc * C2 + c];
      float e0, e1, e2;
      if (c < C) {
        const float* sc = sb + (size_t)c * 3 * N;
        e0 = sc[j] - sc[n];
        e1 = sc[N + j] - sc[N + n];
        e2 = sc[2 * N + j] - sc[2 * N + n];
      } else {
        const float* sc = sb + (size_t)(c - C) * 3 * N;
        e0 = sc[n]; e1 = sc[N + n]; e2 = sc[2 * N + n];
      }
      p0 += wf * e0; p1 += wf * e1; p2 += wf * e2;
      q0 += wd * e0; q1 += wd * e1; q2 += wd * e2;
    }
    vn_bn_leaky(p0, p1, p2, q0, q1, q2, g[oc], be[oc], mu[oc], va[oc]);
    m0 += p0; m1 += p1; m2 += p2;
  }
  const float inv = 1.f / (float)KNN;
  m0 *= inv; m1 *= inv; m2 *= inv;
  size_t ob = ((size_t)b * outChanTot + outChanOfs + oc) * 3 * N + n;
  out[ob] = m0; out[ob + N] = m1; out[ob + 2 * N] = m2;
}

// conv6 shared direction: d6[b][d][n] = sum_c Wd[c] * x123[b][c][d][n]
__global__ void dirshared_kernel(const float* __restrict__ Wd, const float* __restrict__ x123,
                                 float* __restrict__ d6, int total) {
  int tid = blockIdx.x * blockDim.x + threadIdx.x;
  if (tid >= total) return;
  int dn = tid % (3 * NPTS);
  int b  = tid / (3 * NPTS);
  const float* xb = x123 + (size_t)b * 63 * 3 * NPTS;
  float s = 0.f;
  for (int c = 0; c < 63; ++c) s += Wd[c] * xb[(size_t)c * 3 * NPTS + dn];
  d6[(size_t)b * 3 * NPTS + dn] = s;
}

// conv6 epilogue, in-place on p6: BN + leaky with shared direction d6.
__global__ void bnleaky_shared_kernel(float* __restrict__ p6, const float* __restrict__ d6,
                                      const float* __restrict__ g, const float* __restrict__ be,
                                      const float* __restrict__ mu, const float* __restrict__ va,
                                      int total) {
  int tid = blockIdx.x * blockDim.x + threadIdx.x;
  if (tid >= total) return;
  int n  = tid % NPTS;
  int oc = (tid / NPTS) % FDIM;
  int b  = tid / (NPTS * FDIM);
  size_t base = ((size_t)b * FDIM + oc) * 3 * NPTS + n;
  float p0 = p6[base], p1 = p6[base + NPTS], p2 = p6[base + 2 * NPTS];
  size_t db = (size_t)b * 3 * NPTS + n;
  float q0 = d6[db], q1 = d6[db + NPTS], q2 = d6[db + 2 * NPTS];
  vn_bn_leaky(p0, p1, p2, q0, q1, q2, g[oc], be[oc], mu[oc], va[oc]);
  p6[base] = p0; p6[base + NPTS] = p1; p6[base + 2 * NPTS] = p2;
}

// mean over N, duplicated into both halves of hp [B][682][3]
__global__ void meann_kernel(const float* __restrict__ h6, float* __restrict__ hp, int total) {
  int tid = blockIdx.x * blockDim.x + threadIdx.x;
  if (tid >= total) return;
  int d = tid % 3;
  int c = (tid / 3) % FDIM;
  int b = tid / (3 * FDIM);
  const float* hb = h6 + ((size_t)b * FDIM + c) * 3 * NPTS + (size_t)d * NPTS;
  float s = 0.f;
  for (int n = 0; n < NPTS; ++n) s += hb[n];
  s *= (1.f / (float)NPTS);
  hp[((size_t)b * 682 + c) * 3 + d]        = s;
  hp[((size_t)b * 682 + c + FDIM) * 3 + d] = s;
}

// out[b][o][d] = sum_c W[o][c] * in[b][c][d]   (in/out layout [B][C][3])
__global__ void lin_bcd_kernel(const float* __restrict__ W, const float* __restrict__ in,
                               float* __restrict__ out, int O, int C, int total) {
  int tid = blockIdx.x * blockDim.x + threadIdx.x;
  if (tid >= total) return;
  int d = tid % 3;
  int o = (tid / 3) % O;
  int b = tid / (3 * O);
  const float* ib = in + (size_t)b * C * 3;
  float s = 0.f;
  for (int c = 0; c < C; ++c) s += W[(size_t)o * C + c] * ib[c * 3 + d];
  out[((size_t)b * O + o) * 3 + d] = s;
}

// vim = cross(normalize(j), v) per (b, channel)
__global__ void cplx_pre_kernel(const float* __restrict__ v, const float* __restrict__ j,
                                float* __restrict__ vim, int total) {
  int tid = blockIdx.x * blockDim.x + threadIdx.x;
  if (tid >= total) return;
  int c = tid % FDIM;
  int b = tid / FDIM;
  size_t o = ((size_t)b * FDIM + c) * 3;
  float j0 = j[o], j1 = j[o + 1], j2 = j[o + 2];
  float inv = 1.f / (sqrtf(j0 * j0 + j1 * j1 + j2 * j2) + EPSV);
  j0 *= inv; j1 *= inv; j2 *= inv;
  float v0 = v[o], v1 = v[o + 1], v2 = v[o + 2];
  vim[o]     = j1 * v2 - j2 * v1;
  vim[o + 1] = j2 * v0 - j0 * v2;
  vim[o + 2] = j0 * v1 - j1 * v0;
}

// y = leaky( Wre*v + Wim*vim ; dir = Wdir*v )
__global__ void cplx_main_kernel(const float* __restrict__ Wre, const float* __restrict__ Wim,
                                 const float* __restrict__ Wdir,
                                 const float* __restrict__ v, const float* __restrict__ vim,
                                 float* __restrict__ y, int total) {
  int tid = blockIdx.x * blockDim.x + threadIdx.x;
  if (tid >= total) return;
  int oc = tid % FDIM;
  int b  = tid / FDIM;
  const float* vb = v   + (size_t)b * FDIM * 3;
  const float* ib = vim + (size_t)b * FDIM * 3;
  float p0 = 0, p1 = 0, p2 = 0, q0 = 0, q1 = 0, q2 = 0;
  for (int c = 0; c < FDIM; ++c) {
    float wre = Wre[(size_t)oc * FDIM + c];
    float wim = Wim[(size_t)oc * FDIM + c];
    float wdr = Wdir[(size_t)oc * FDIM + c];
    float a0 = vb[c * 3], a1 = vb[c * 3 + 1], a2 = vb[c * 3 + 2];
    float m0 = ib[c * 3], m1 = ib[c * 3 + 1], m2 = ib[c * 3 + 2];
    p0 += wre * a0 + wim * m0;
    p1 += wre * a1 + wim * m1;
    p2 += wre * a2 + wim * m2;
    q0 += wdr * a0; q1 += wdr * a1; q2 += wdr * a2;
  }
  vn_leaky_only(p0, p1, p2, q0, q1, q2);
  size_t o = ((size_t)b * FDIM + oc) * 3;
  y[o] = p0; y[o + 1] = p1; y[o + 2] = p2;
}

// dst [B][2*Chalf][3] = concat(srcA, srcB) along channels
__global__ void concat_half_kernel(const float* __restrict__ a, const float* __restrict__ bsrc,
                                   float* __restrict__ dst, int Chalf, int total) {
  int tid = blockIdx.x * blockDim.x + threadIdx.x;
  if (tid >= total) return;
  int C2 = 2 * Chalf;
  int d = tid % 3;
  int c = (tid / 3) % C2;
  int b = tid / (3 * C2);
  float val = (c < Chalf) ? a[((size_t)b * Chalf + c) * 3 + d]
                          : bsrc[((size_t)b * Chalf + (c - Chalf)) * 3 + d];
  dst[((size_t)b * C2 + c) * 3 + d] = val;
}

// VN linear + BN + leaky on [B][C][3] -> [B][O][3] (std_vn1/std_vn2)
__global__ void vnblock_vec_kernel(const float* __restrict__ Wf, const float* __restrict__ Wd,
                                   const float* __restrict__ g, const float* __restrict__ be,
                                   const float* __restrict__ mu, const float* __restrict__ va,
                                   const float* __restrict__ in, float* __restrict__ out,
                                   int O, int C, int total) {
  int tid = blockIdx.x * blockDim.x + threadIdx.x;
  if (tid >= total) return;
  int oc = tid % O;
  int b  = tid / O;
  const float* ib = in + (size_t)b * C * 3;
  float p0 = 0, p1 = 0, p2 = 0, q0 = 0, q1 = 0, q2 = 0;
  for (int c = 0; c < C; ++c) {
    float wf = Wf[(size_t)oc * C + c], wd = Wd[(size_t)oc * C + c];
    float a0 = ib[c * 3], a1 = ib[c * 3 + 1], a2 = ib[c * 3 + 2];
    p0 += wf * a0; p1 += wf * a1; p2 += wf * a2;
    q0 += wd * a0; q1 += wd * a1; q2 += wd * a2;
  }
  vn_bn_leaky(p0, p1, p2, q0, q1, q2, g[oc], be[oc], mu[oc], va[oc]);
  size_t o = ((size_t)b * O + oc) * 3;
  out[o] = p0; out[o + 1] = p1; out[o + 2] = p2;
}

// x_std[b][c][k] = sum_j xout[b][c][j] * z0[b][k][j]   (z0 stored [B][3][3] as (o,d))
__global__ void xstd_kernel(const float* __restrict__ xout, const float* __restrict__ z0,
                            float* __restrict__ xstd, int total) {
  int tid = blockIdx.x * blockDim.x + threadIdx.x;
  if (tid >= total) return;
  int k = tid % 3;
  int c = (tid / 3) % 682;
  int b = tid / (3 * 682);
  const float* xb = xout + ((size_t)b * 682 + c) * 3;
  const float* zb = z0 + (size_t)b * 9;
  float s = xb[0] * zb[k * 3 + 0] + xb[1] * zb[k * 3 + 1] + xb[2] * zb[k * 3 + 2];
  xstd[((size_t)b * 682 + c) * 3 + k] = s;
}

// x1_out[b][c][o] = sum_d W[o][d] * xstd[b][c][d] + bias[o]
__global__ void lin0_kernel(const float* __restrict__ W, const float* __restrict__ bias,
                            const float* __restrict__ xstd, float* __restrict__ out, int total) {
  int tid = blockIdx.x * blockDim.x + threadIdx.x;
  if (tid >= total) return;
  int o = tid % 1024;
  int c = (tid / 1024) % 682;
  int b = tid / (1024 * 682);
  const float* xb = xstd + ((size_t)b * 682 + c) * 3;
  float s = W[o * 3 + 0] * xb[0] + W[o * 3 + 1] * xb[1] + W[o * 3 + 2] * xb[2] + bias[o];
  out[((size_t)b * 682 + c) * 1024 + o] = s;
}

// ---------------------------------------------------------------------------
// Host launch
// ---------------------------------------------------------------------------
extern "C" void kernel_launch(void* const* d_in, const int* in_sizes, int n_in,
                              void* d_out, int out_size, void* d_ws, size_t ws_size,
                              hipStream_t stream) {
  (void)in_sizes; (void)n_in; (void)out_size; (void)ws_size;
  const int N = NPTS, B = NB;
#define FP(i) ((const float*)d_in[i])
  const float* x = FP(0);
  // param flat order (depth-first dict order):
  // conv1=1..6  conv2=7..12  conv3=13..18  conv4=19..24  conv5=25..30
  // conv6=31..36  lin1..4=37..40  cplx1=41..43  cplx2=44..46
  // std_vn1=47..52  std_vn2=53..58  std_lin=59  lin0_W=60  lin0_b=61

  // workspace carve-up
  uintptr_t wsbase = (uintptr_t)d_ws;
  size_t off = 0;
  auto carve = [&](size_t bytes) -> void* {
    off = (off + 255) & ~(size_t)255;
    void* p = (void*)(wsbase + off);
    off += bytes;
    return p;
  };
  int*   idxb  = (int*)  carve((size_t)B * N * KNN * sizeof(int));
  float* hA    = (float*)carve((size_t)B * 21 * 3 * N * KNN * sizeof(float));
  float* x123  = (float*)carve((size_t)B * 63 * 3 * N * sizeof(float));
  float* p6    = (float*)carve((size_t)B * FDIM * 3 * N * sizeof(float));  // also h6 (in-place)
  float* d6    = (float*)carve((size_t)B * 3 * N * sizeof(float));
  float* hp    = (float*)carve((size_t)B * 682 * 3 * sizeof(float));
  float* vb    = (float*)carve((size_t)B * FDIM * 3 * sizeof(float));
  float* jb    = (float*)carve((size_t)B * FDIM * 3 * sizeof(float));
  float* vimb  = (float*)carve((size_t)B * FDIM * 3 * sizeof(float));
  float* yb    = (float*)carve((size_t)B * FDIM * 3 * sizeof(float));
  float* comb1 = (float*)carve((size_t)B * 682 * 3 * sizeof(float));
  float* xb2   = (float*)carve((size_t)B * FDIM * 3 * sizeof(float));
  float* j2b   = (float*)carve((size_t)B * FDIM * 3 * sizeof(float));
  float* y2b   = (float*)carve((size_t)B * FDIM * 3 * sizeof(float));
  float* comb2 = (float*)carve((size_t)B * 682 * 3 * sizeof(float));      // = xout
  float* z1    = (float*)carve((size_t)B * FDIM * 3 * sizeof(float));
  float* z2    = (float*)carve((size_t)B * 170 * 3 * sizeof(float));
  float* z0    = (float*)carve((size_t)B * 9 * sizeof(float));
  float* xstd  = (float*)carve((size_t)B * 682 * 3 * sizeof(float));

  dim3 knngrid(N / 16, B);

  // ---- stage 1: knn on raw xyz (D=3), conv1, conv2+mean -> x1 (x123 ch 0..20)
  knn_wmma_kernel<3><<<knngrid, 32, 0, stream>>>(x, 3 * N, idxb);
  {
    int total = B * N * KNN * 21;
    vn_edgeconv_kernel<<<(total + 255) / 256, 256, 0, stream>>>(
        x, 3 * N, 1, idxb, FP(1), FP(2), FP(3), FP(4), FP(5), FP(6), 21, hA, total);
  }
  {
    int total = B * N * 21;
    vn_conv_mean_kernel<<<(total + 255) / 256, 256, 0, stream>>>(
        hA, 21, FP(7), FP(8), FP(9), FP(10), FP(11), FP(12), 21, x123, 0, 63, total);
  }
  // ---- stage 2: knn on x1 (D=63), conv3, conv4+mean -> x2 (x123 ch 21..41)
  knn_wmma_kernel<63><<<knngrid, 32, 0, stream>>>(x123, 63 * 3 * N, idxb);
  {
    int total = B * N * KNN * 21;
    vn_edgeconv_kernel<<<(total + 255) / 256, 256, 0, stream>>>(
        x123, 63 * 3 * N, 21, idxb, FP(13), FP(14), FP(15), FP(16), FP(17), FP(18), 21, hA, total);
  }
  {
    int total = B * N * 21;
    vn_conv_mean_kernel<<<(total + 255) / 256, 256, 0, stream>>>(
        hA, 21, FP(19), FP(20), FP(21), FP(22), FP(23), FP(24), 21, x123, 21, 63, total);
  }
  // ---- stage 3: knn on x2 (D=63), conv5+mean -> x3 (x123 ch 42..62)
  knn_wmma_kernel<63><<<knngrid, 32, 0, stream>>>(x123 + (size_t)21 * 3 * N, 63 * 3 * N, idxb);
  {
    int total = B * N * 21;
    vn_edgeconv_mean_kernel<<<(total + 255) / 256, 256, 0, stream>>>(
        x123 + (size_t)21 * 3 * N, 63 * 3 * N, 21, idxb,
        FP(25), FP(26), FP(27), FP(28), FP(29), FP(30), 21, x123, 42, 63, total);
  }
  // ---- conv6: WMMA GEMM [341x63] x [63 x 3N] per batch (4 col tiles/wave),
  //      then shared-dir BN+leaky in place.
  gemm_wmma_kernel<63, 4><<<dim3((FDIM + 15) / 16, (3 * N) / (16 * 4), B), 32, 0, stream>>>(
      FP(31), x123, p6, FDIM, 3 * N);
  {
    int total = B * 3 * N;
    dirshared_kernel<<<(total + 255) / 256, 256, 0, stream>>>(FP(32), x123, d6, total);
  }
  {
    int total = B * FDIM * N;
    bnleaky_shared_kernel<<<(total + 255) / 256, 256, 0, stream>>>(
        p6, d6, FP(33), FP(34), FP(35), FP(36), total);
  }
  {
    int total = B * FDIM * 3;
    meann_kernel<<<(total + 255) / 256, 256, 0, stream>>>(p6, hp, total);
  }
  // ---- lin1/lin2 -> v, j ; complex block 1
  {
    int total = B * FDIM * 3;
    lin_bcd_kernel<<<(total + 255) / 256, 256, 0, stream>>>(FP(37), hp, vb, FDIM, 682, total);
    lin_bcd_kernel<<<(total + 255) / 256, 256, 0, stream>>>(FP(38), hp, jb, FDIM, 682, total);
  }
  {
    int total = B * FDIM;
    cplx_pre_kernel<<<(total + 255) / 256, 256, 0, stream>>>(vb, jb, vimb, total);
    cplx_main_kernel<<<(total + 255) / 256, 256, 0, stream>>>(FP(41), FP(42), FP(43), vb, vimb, yb, total);
  }
  {
    int total = B * 682 * 3;
    concat_half_kernel<<<(total + 255) / 256, 256, 0, stream>>>(vb, yb, comb1, FDIM, total);
  }
  // ---- lin3/lin4 ; complex block 2 -> xout (comb2)
  {
    int total = B * FDIM * 3;
    lin_bcd_kernel<<<(total + 255) / 256, 256, 0, stream>>>(FP(39), comb1, xb2, FDIM, 682, total);
    lin_bcd_kernel<<<(total + 255) / 256, 256, 0, stream>>>(FP(40), comb1, j2b, FDIM, 682, total);
  }
  {
    int total = B * FDIM;
    cplx_pre_kernel<<<(total + 255) / 256, 256, 0, stream>>>(xb2, j2b, vimb, total);
    cplx_main_kernel<<<(total + 255) / 256, 256, 0, stream>>>(FP(44), FP(45), FP(46), xb2, vimb, y2b, total);
  }
  {
    int total = B * 682 * 3;
    concat_half_kernel<<<(total + 255) / 256, 256, 0, stream>>>(xb2, y2b, comb2, FDIM, total);
  }
  // ---- VNStdFeature
  {
    int total = B * FDIM;
    vnblock_vec_kernel<<<(total + 255) / 256, 256, 0, stream>>>(
        FP(47), FP(48), FP(49), FP(50), FP(51), FP(52), comb2, z1, FDIM, 682, total);
  }
  {
    int total = B * 170;
    vnblock_vec_kernel<<<(total + 255) / 256, 256, 0, stream>>>(
        FP(53), FP(54), FP(55), FP(56), FP(57), FP(58), z1, z2, 170, FDIM, total);
  }
  {
    int total = B * 3 * 3;
    lin_bcd_kernel<<<(total + 255) / 256, 256, 0, stream>>>(FP(59), z2, z0, 3, 170, total);
  }
  {
    int total = B * 682 * 3;
    xstd_kernel<<<(total + 255) / 256, 256, 0, stream>>>(comb2, z0, xstd, total);
  }
  // ---- outputs: xout first, then x1_out
  hipMemcpyAsync(d_out, comb2, (size_t)B * 682 * 3 * sizeof(float),
                 hipMemcpyDeviceToDevice, stream);
  {
    int total = B * 682 * 1024;
    lin0_kernel<<<(total + 255) / 256, 256, 0, stream>>>(
        FP(60), FP(61), xstd, (float*)d_out + (size_t)B * 682 * 3, total);
  }
#undef FP
}